// AdaDualFocal_22574348108420
// MI455X (gfx1250) — compile-verified
//
#include <hip/hip_runtime.h>
#include <math.h>
#include <stdint.h>

// AdaDualFocal loss for MI455X (gfx1250):
//   N=4096 rows, C=32000 classes, 15 calibration bins.
// Strategy: one 32-wave workgroup per row. Stage the 128KB row into LDS with
// gfx1250 async global->LDS B128 copies (ASYNCcnt), then:
//   pass 1 (LDS): rowmax and m2 = max{x_i : x_i < x_k}
//   pass 2 (LDS): sum exp(x - rowmax)
// Thread 0 finishes the scalar math; a second kernel does a deterministic
// tree-sum of the 4096 per-row losses.

#define N_ROWS    4096
#define C_CLASSES 32000
#define NBINS     15
#define BLOCK     1024
#define WAVES     (BLOCK / 32)
#define CHUNKS    (C_CLASSES / 4)   // 8000 float4 chunks per row

// ---- gfx1250 async global->LDS copy (16 bytes per lane) ----------------
__device__ __forceinline__ void async_copy_b128(uint32_t lds_byte_off,
                                                const float* gptr) {
#if defined(__gfx1250__)
    // VDST = LDS byte address, VADDR = 64-bit global address, SADDR = off
    asm volatile("global_load_async_to_lds_b128 %0, %1, off"
                 :
                 : "v"(lds_byte_off), "v"(gptr)
                 : "memory");
#else
    (void)lds_byte_off; (void)gptr;
#endif
}

__device__ __forceinline__ void wait_async_zero() {
#if defined(__gfx1250__)
#if defined(__has_builtin)
#if __has_builtin(__builtin_amdgcn_s_wait_asynccnt)
    __builtin_amdgcn_s_wait_asynccnt(0);
#else
    asm volatile("s_wait_asynccnt 0x0" ::: "memory");
#endif
#else
    asm volatile("s_wait_asynccnt 0x0" ::: "memory");
#endif
#endif
}

__device__ __forceinline__ float wave_max(float v) {
    #pragma unroll
    for (int o = 16; o > 0; o >>= 1) v = fmaxf(v, __shfl_xor(v, o, 32));
    return v;
}
__device__ __forceinline__ float wave_sum(float v) {
    #pragma unroll
    for (int o = 16; o > 0; o >>= 1) v += __shfl_xor(v, o, 32);
    return v;
}

__global__ __launch_bounds__(BLOCK)
void ada_dual_focal_row_kernel(const float* __restrict__ input,
                               const int*   __restrict__ target,
                               const float* __restrict__ bin_uppers,
                               const float* __restrict__ bin_gammas,
                               float*       __restrict__ row_loss) {
    __shared__ __align__(16) float lds_row[C_CLASSES];
    __shared__ float red_a[WAVES];
    __shared__ float red_b[WAVES];
    __shared__ float s_bcast[4];   // [0]=rowmax [1]=m2

    const int row = blockIdx.x;
    const int tid = threadIdx.x;
    const int lane = tid & 31;
    const int wid  = tid >> 5;

    const float* grow = input + (size_t)row * C_CLASSES;

    // ---- stage row to LDS via async copies (HBM read exactly once) ----
    const uint32_t lds_base = (uint32_t)(uintptr_t)&lds_row[0];
    #pragma unroll
    for (int i = 0; i < 8; ++i) {
        int idx = tid + i * BLOCK;
        if (idx < CHUNKS) {
            async_copy_b128(lds_base + (uint32_t)idx * 16u, grow + (size_t)idx * 4);
        }
    }
    wait_async_zero();     // this wave's async copies done
    __syncthreads();       // all waves' copies done -> LDS row valid

    const float xk = lds_row[target[row]];

    // ---- pass 1: rowmax and m2 = max of elements strictly below xk ----
    float vmax = -__builtin_inff();
    float vm2  = -__builtin_inff();
    const float4* lrow4 = (const float4*)lds_row;
    #pragma unroll
    for (int i = 0; i < 8; ++i) {
        int idx = tid + i * BLOCK;
        if (idx < CHUNKS) {
            float4 v = lrow4[idx];
            vmax = fmaxf(vmax, fmaxf(fmaxf(v.x, v.y), fmaxf(v.z, v.w)));
            vm2 = fmaxf(vm2, (v.x < xk) ? v.x : -__builtin_inff());
            vm2 = fmaxf(vm2, (v.y < xk) ? v.y : -__builtin_inff());
            vm2 = fmaxf(vm2, (v.z < xk) ? v.z : -__builtin_inff());
            vm2 = fmaxf(vm2, (v.w < xk) ? v.w : -__builtin_inff());
        }
    }
    vmax = wave_max(vmax);
    vm2  = wave_max(vm2);
    if (lane == 0) { red_a[wid] = vmax; red_b[wid] = vm2; }
    __syncthreads();
    if (tid < 32) {
        float a = wave_max(red_a[tid]);
        float b = wave_max(red_b[tid]);
        if (tid == 0) { s_bcast[0] = a; s_bcast[1] = b; }
    }
    __syncthreads();
    const float rowmax = s_bcast[0];

    // ---- pass 2: sum of exp(x - rowmax), from LDS ----
    float psum = 0.0f;
    #pragma unroll
    for (int i = 0; i < 8; ++i) {
        int idx = tid + i * BLOCK;
        if (idx < CHUNKS) {
            float4 v = lrow4[idx];
            psum += __expf(v.x - rowmax);
            psum += __expf(v.y - rowmax);
            psum += __expf(v.z - rowmax);
            psum += __expf(v.w - rowmax);
        }
    }
    psum = wave_sum(psum);
    if (lane == 0) red_a[wid] = psum;
    __syncthreads();
    if (tid < 32) {
        float s = wave_sum(red_a[tid]);
        if (tid == 0) {
            const float m2   = s_bcast[1];
            const float lse  = logf(s);
            const float lpk  = xk - rowmax - lse;          // logp_k
            const float p_k  = expf(lpk);
            const float p_j  = (m2 == -__builtin_inff()) ? 0.0f
                                                         : expf(m2 - rowmax - lse);
            const float pt   = p_k - p_j;
            int bidx = 0;
            #pragma unroll
            for (int b = 0; b < NBINS; ++b) bidx += (pt >= bin_uppers[b]) ? 1 : 0;
            if (bidx > NBINS - 1) bidx = NBINS - 1;
            const float gamma = bin_gammas[bidx];
            row_loss[row] = -powf(1.0f - pt, gamma) * lpk;
        }
    }
}

// Deterministic tree reduction of the 4096 per-row losses into d_out[0].
__global__ __launch_bounds__(1024)
void ada_dual_focal_sum_kernel(const float* __restrict__ row_loss,
                               float* __restrict__ out) {
    __shared__ float red[32];
    float s = 0.0f;
    #pragma unroll
    for (int i = 0; i < N_ROWS / 1024; ++i) s += row_loss[threadIdx.x + i * 1024];
    s = wave_sum(s);
    if ((threadIdx.x & 31) == 0) red[threadIdx.x >> 5] = s;
    __syncthreads();
    if (threadIdx.x < 32) {
        float t = wave_sum(red[threadIdx.x]);
        if (threadIdx.x == 0) out[0] = t;
    }
}

extern "C" void kernel_launch(void* const* d_in, const int* in_sizes, int n_in,
                              void* d_out, int out_size, void* d_ws, size_t ws_size,
                              hipStream_t stream) {
    const float* input      = (const float*)d_in[0];
    const int*   target     = (const int*)d_in[1];
    const float* bin_uppers = (const float*)d_in[2];
    const float* bin_gammas = (const float*)d_in[3];

    float* row_loss = (float*)d_ws;   // 4096 floats of scratch

    ada_dual_focal_row_kernel<<<N_ROWS, BLOCK, 0, stream>>>(
        input, target, bin_uppers, bin_gammas, row_loss);
    ada_dual_focal_sum_kernel<<<1, 1024, 0, stream>>>(row_loss, (float*)d_out);
}